// CTCLoss_4123168604400
// MI455X (gfx1250) — compile-verified
//
#include <hip/hip_runtime.h>
#include <hip/hip_bf16.h>
#include <stdint.h>

// Problem constants from the reference
#define TT 128          // time steps
#define SS 16           // target length
#define CC 128          // classes
#define LEXT 33         // 2*SS+1 extended label states
#define T_CHUNK 64      // rows per TDM tile (32KB f32 tile in LDS)
#define NEGV (-1e30f)

typedef __attribute__((ext_vector_type(4))) unsigned int u32x4;
typedef __attribute__((ext_vector_type(8))) int          i32x8;
typedef __attribute__((ext_vector_type(4))) int          i32x4;

__device__ __forceinline__ float lae(float a, float b) {
    // logaddexp, matches jnp.logaddexp numerics with NEG sentinel
    float m = fmaxf(a, b);
    float d = fabsf(a - b);
    return m + log1pf(__expf(-d));
}

__global__ __launch_bounds__(64) void ctc_dp_kernel(
    const float* __restrict__ pred,   // (B, T, C) log-probs
    const float* __restrict__ lab,    // (B, S, C)
    float* __restrict__ ws)           // (B,) per-batch nll/S
{
    __shared__ float tile[T_CHUNK * CC];   // 32 KB TDM destination
    __shared__ int   targ[SS];
    __shared__ int   extc[LEXT];
    __shared__ int   skipok[LEXT];
    __shared__ float bufA[LEXT + 2];       // 2 leading NEG pads
    __shared__ float bufB[LEXT + 2];

    const int b   = blockIdx.x;
    const int tid = threadIdx.x;

    // ---- targets[s] = argmax_c lable[b,s,c] (first max wins, like jnp.argmax)
    if (tid < SS) {
        const float* row = lab + ((size_t)b * SS + tid) * CC;
        float best = row[0];
        int   bi   = 0;
        for (int c = 1; c < CC; ++c) {
            float v = row[c];
            if (v > best) { best = v; bi = c; }
        }
        targ[tid] = bi;
    }
    __syncthreads();

    // ---- extended label + skip-transition mask
    if (tid < LEXT) {
        extc[tid] = (tid & 1) ? targ[tid >> 1] : 0;  // blanks at even slots
        int sk = 0;
        if (tid >= 2 && (tid & 1))
            sk = (targ[tid >> 1] != targ[(tid >> 1) - 1]) ? 1 : 0;
        skipok[tid] = sk;
    }
    if (tid < 2) { bufA[tid] = NEGV; bufB[tid] = NEGV; }
    __syncthreads();

    float* cur = bufA;
    float* nxt = bufB;
    const unsigned lds_tile = (unsigned)(uintptr_t)(&tile[0]); // low 32 bits = LDS byte offset

    for (int chunk = 0; chunk < TT / T_CHUNK; ++chunk) {
        // ---- TDM: DMA a (T_CHUNK x CC) f32 tile Global->LDS.
        // Branch is wave-uniform: only wave 0 issues (TDM ignores EXEC, issues per wave).
        if (tid < 32) {
            uint64_t gaddr = (uint64_t)(uintptr_t)
                (pred + ((size_t)b * TT + (size_t)chunk * T_CHUNK) * CC);
            // D# group 0: count=1 | lds_addr | global_addr[56:0] | type=2
            u32x4 g0 = {
                1u,
                lds_tile,
                (unsigned)(gaddr & 0xFFFFFFFFu),
                ((unsigned)((gaddr >> 32) & 0x01FFFFFFu)) | (2u << 30)
            };
            // D# group 1: data_size=4B; tensor 64x128 f32; tile 64x128; dim0 stride=128
            i32x8 g1 = {
                (int)(2u << 16),          // [17:16] data_size = 4B; wg_mask=0
                (int)(CC << 16),          // tensor_dim0 lo16 = 128  (bits 63:48)
                (int)(T_CHUNK << 16),     // dim0 hi=0 | tensor_dim1 lo16 = 64
                (int)(CC << 16),          // dim1 hi=0 | tile_dim0 = 128
                (int)T_CHUNK,             // tile_dim1 = 64, tile_dim2 = 0
                (int)CC,                  // tensor_dim0_stride = 128 elements
                0,                        // stride0 hi | tensor_dim1_stride lo (unused, 2D)
                0
            };
            i32x4 gz4 = {0, 0, 0, 0};              // groups 2/3: dims beyond 2D unused
            i32x8 gz8 = {0, 0, 0, 0, 0, 0, 0, 0};  // extra group (clang-23 6-arg form), unused
            __builtin_amdgcn_tensor_load_to_lds(g0, g1, gz4, gz4, gz8, 0);
            __builtin_amdgcn_s_wait_tensorcnt(0);
        }
        __syncthreads();   // publish TDM-written LDS to wave 1

        int tt0 = 0;
        if (chunk == 0) {
            // alpha_0: only states 0 (blank) and 1 (first label) live
            if (tid < LEXT)
                cur[tid + 2] = (tid < 2) ? tile[extc[tid]] : NEGV;
            __syncthreads();
            tt0 = 1;
        }

        for (int tt = tt0; tt < T_CHUNK; ++tt) {
            if (tid < LEXT) {
                float a1 = cur[tid + 2];                     // stay
                float a2 = cur[tid + 1];                     // from l-1
                float a3 = skipok[tid] ? cur[tid] : NEGV;    // from l-2
                float lp = tile[tt * CC + extc[tid]];
                nxt[tid + 2] = lae(lae(a1, a2), a3) + lp;
            }
            __syncthreads();
            float* tmp = cur; cur = nxt; nxt = tmp;
        }
        __syncthreads();   // all tile reads done before next chunk's DMA overwrites it
    }

    if (tid == 0) {
        float res = lae(cur[2 + LEXT - 1], cur[2 + LEXT - 2]);
        float nll = -res;
        if (!isfinite(nll)) nll = 0.0f;   // zero_infinity
        ws[b] = nll / (float)SS;          // reduction pre-scale by target length
    }
}

// Deterministic final mean (no float atomics: bit-identical across replays)
__global__ __launch_bounds__(1024) void ctc_reduce_kernel(
    const float* __restrict__ ws, float* __restrict__ out, int n)
{
    __shared__ float sh[1024];
    const int tid = threadIdx.x;
    float s = 0.0f;
    for (int i = tid; i < n; i += 1024) s += ws[i];
    sh[tid] = s;
    __syncthreads();
    for (int off = 512; off > 0; off >>= 1) {
        if (tid < off) sh[tid] += sh[tid + off];
        __syncthreads();
    }
    if (tid == 0) out[0] = sh[0] / (float)n;
}

extern "C" void kernel_launch(void* const* d_in, const int* in_sizes, int n_in,
                              void* d_out, int out_size, void* d_ws, size_t ws_size,
                              hipStream_t stream) {
    const float* pred = (const float*)d_in[0];   // (B, T, C) f32
    const float* lab  = (const float*)d_in[1];   // (B, S, C) f32
    float* out = (float*)d_out;                  // scalar f32
    float* ws  = (float*)d_ws;                   // B floats of scratch

    const int B = in_sizes[0] / (TT * CC);       // 4096

    ctc_dp_kernel<<<B, 64, 0, stream>>>(pred, lab, ws);
    ctc_reduce_kernel<<<1, 1024, 0, stream>>>(ws, out, B);
}